// CREStereo_88467736363602
// MI455X (gfx1250) — compile-verified
//
#include <hip/hip_runtime.h>
#include <hip/hip_bf16.h>
#include <math.h>

// ---------------- problem constants (from setup_inputs) ----------------
#define NN 2
#define CC 256
#define HH 128
#define WW 240
#define SS 9
#define CG 64            // channels per group
#define TW 16            // x-tile width for WMMA gram

// d_out layout (floats): corrs_offset | corrs_iter | flow_up
#define OUT_OFF0 ((size_t)0)
#define OUT_OFF1 ((size_t)NN * 36 * HH * WW)                  // 2,211,840
#define OUT_OFF2 (OUT_OFF1 + (size_t)NN * 36 * HH * WW)       // 4,423,680

typedef __attribute__((ext_vector_type(2))) float v2f;
typedef __attribute__((ext_vector_type(8))) float v8f;

// =====================================================================
// Kernel A: flow_up = -convex_upsample(flow, mask, rate=4)
// =====================================================================
__global__ void crestereo_upsample_kernel(const float* __restrict__ flow,
                                          const float* __restrict__ mask,
                                          float* __restrict__ out)
{
    int idx = blockIdx.x * blockDim.x + threadIdx.x;
    if (idx >= NN * HH * WW) return;
    int x = idx % WW;
    int y = (idx / WW) % HH;
    int n = idx / (WW * HH);

    float nbx[9], nby[9];
#pragma unroll
    for (int k = 0; k < 9; k++) {
        int dy = k / 3 - 1, dx = k % 3 - 1;
        int ny = y + dy, nx = x + dx;
        bool ok = (ny >= 0) && (ny < HH) && (nx >= 0) && (nx < WW);
        size_t b0 = (((size_t)n * 2 + 0) * HH + ny) * WW + nx;
        size_t b1 = (((size_t)n * 2 + 1) * HH + ny) * WW + nx;
        nbx[k] = ok ? 4.0f * flow[b0] : 0.0f;
        nby[k] = ok ? 4.0f * flow[b1] : 0.0f;
    }

    float* up = out + OUT_OFF2;
#pragma unroll 1
    for (int r = 0; r < 16; r++) {          // r = r1*4 + r2
        float mv[9];
        float mx = -1e30f;
#pragma unroll
        for (int k = 0; k < 9; k++) {
            mv[k] = mask[(((size_t)n * 144 + k * 16 + r) * HH + y) * WW + x];
            mx = fmaxf(mx, mv[k]);
        }
        float sum = 0.0f;
#pragma unroll
        for (int k = 0; k < 9; k++) { mv[k] = __expf(mv[k] - mx); sum += mv[k]; }
        float inv = 1.0f / sum;
        float ux = 0.0f, uy = 0.0f;
#pragma unroll
        for (int k = 0; k < 9; k++) { ux += mv[k] * nbx[k]; uy += mv[k] * nby[k]; }
        ux *= inv; uy *= inv;
        int r1 = r >> 2, r2 = r & 3;
        size_t oy = (size_t)4 * y + r1;
        size_t ox = (size_t)4 * x + r2;
        up[(((size_t)n * 2 + 0) * (4 * HH) + oy) * (4 * WW) + ox] = -ux;
        up[(((size_t)n * 2 + 1) * (4 * HH) + oy) * (4 * WW) + ox] = -uy;
    }
}

// =====================================================================
// Kernel B: corr_att_offset (psize = (1,9))
// one thread per (n, s, y, x); tap offsets/weights hoisted out of the
// 256-channel loop; x fastest -> coalesced gathers
// =====================================================================
__global__ void crestereo_corr_att_kernel(const float* __restrict__ fmap1,
                                          const float* __restrict__ fmap2,
                                          const float* __restrict__ flow,
                                          const float* __restrict__ eo,
                                          float* __restrict__ out)
{
    int idx = blockIdx.x * blockDim.x + threadIdx.x;
    if (idx >= NN * SS * HH * WW) return;
    int x = idx % WW;
    int y = (idx / WW) % HH;
    int s = (idx / (WW * HH)) % SS;
    int n = idx / (WW * HH * SS);

    float fx = flow[(((size_t)n * 2 + 0) * HH + y) * WW + x];
    float fy = flow[(((size_t)n * 2 + 1) * HH + y) * WW + x];
    float ex = eo[(((size_t)n * (2 * SS) + 2 * s + 0) * HH + y) * WW + x];
    float ey = eo[(((size_t)n * (2 * SS) + 2 * s + 1) * HH + y) * WW + x];

    // offs for psize (1,9): (s-4, 0)
    float xs = (float)x + fx + (float)(s - 4) + ex;
    float ys = (float)y + fy + ey;

    float xf = floorf(xs), yf = floorf(ys);
    int ix = (int)xf, iy = (int)yf;
    float wx1 = xs - xf, wy1 = ys - yf;
    float wx0 = 1.0f - wx1, wy0 = 1.0f - wy1;
    float vx0 = (ix >= 0 && ix <= WW - 1) ? 1.0f : 0.0f;
    float vx1 = (ix + 1 >= 0 && ix + 1 <= WW - 1) ? 1.0f : 0.0f;
    float vy0 = (iy >= 0 && iy <= HH - 1) ? 1.0f : 0.0f;
    float vy1 = (iy + 1 >= 0 && iy + 1 <= HH - 1) ? 1.0f : 0.0f;
    int ix0 = min(max(ix, 0), WW - 1),     ix1 = min(max(ix + 1, 0), WW - 1);
    int iy0 = min(max(iy, 0), HH - 1),     iy1 = min(max(iy + 1, 0), HH - 1);

    int   o0 = iy0 * WW + ix0,  o1 = iy0 * WW + ix1;
    int   o2 = iy1 * WW + ix0,  o3 = iy1 * WW + ix1;
    float w0 = wy0 * wx0 * vy0 * vx0, w1 = wy0 * wx1 * vy0 * vx1;
    float w2 = wy1 * wx0 * vy1 * vx0, w3 = wy1 * wx1 * vy1 * vx1;

    const float* f1 = fmap1 + (size_t)n * CC * HH * WW + (size_t)y * WW + x;
    const float* f2 = fmap2 + (size_t)n * CC * HH * WW;

    float acc[4] = {0.f, 0.f, 0.f, 0.f};
#pragma unroll 4
    for (int c = 0; c < CC; c++) {
        const float* p = f2 + (size_t)c * (HH * WW);
        float sv = p[o0] * w0 + p[o1] * w1 + p[o2] * w2 + p[o3] * w3;
        float lv = f1[(size_t)c * (HH * WW)];
        acc[c >> 6] += lv * sv;
    }
#pragma unroll
    for (int g = 0; g < 4; g++)
        out[(((size_t)n * 36 + g * SS + s) * HH + y) * WW + x] = acc[g] * (1.0f / 64.0f);
}

// ---- helper: compute clamped bilinear taps for pixel (yc, xc) -----------
__device__ __forceinline__ void bilin_setup(const float* __restrict__ flow,
                                            int n, int yc, int xc,
                                            int& o0, int& o1, int& o2, int& o3,
                                            float& w0, float& w1, float& w2, float& w3)
{
    float fx = flow[(((size_t)n * 2 + 0) * HH + yc) * WW + xc];
    float fy = flow[(((size_t)n * 2 + 1) * HH + yc) * WW + xc];
    float xs = (float)xc + fx, ys = (float)yc + fy;
    float xf = floorf(xs), yf = floorf(ys);
    int ix = (int)xf, iy = (int)yf;
    float wx1 = xs - xf, wy1 = ys - yf;
    float wx0 = 1.0f - wx1, wy0 = 1.0f - wy1;
    float vx0 = (ix >= 0 && ix <= WW - 1) ? 1.0f : 0.0f;
    float vx1 = (ix + 1 >= 0 && ix + 1 <= WW - 1) ? 1.0f : 0.0f;
    float vy0 = (iy >= 0 && iy <= HH - 1) ? 1.0f : 0.0f;
    float vy1 = (iy + 1 >= 0 && iy + 1 <= HH - 1) ? 1.0f : 0.0f;
    int ix0 = min(max(ix, 0), WW - 1),  ix1 = min(max(ix + 1, 0), WW - 1);
    int iy0 = min(max(iy, 0), HH - 1),  iy1 = min(max(iy + 1, 0), HH - 1);
    o0 = iy0 * WW + ix0;  w0 = wy0 * wx0 * vy0 * vx0;
    o1 = iy0 * WW + ix1;  w1 = wy0 * wx1 * vy0 * vx1;
    o2 = iy1 * WW + ix0;  w2 = wy1 * wx0 * vy1 * vx0;
    o3 = iy1 * WW + ix1;  w3 = wy1 * wx1 * vy1 * vx1;
}

// =====================================================================
// Kernel C: corr_iter (psize = (3,3)) via V_WMMA_F32_16X16X4_F32
// 1 wave per (n, y, 16-px x-tile). Straight-line pixel ownership (no
// indexed VGPR arrays -> no v_movrels in the hot loop). Halo fixups use
// lane shuffles of the already-loaded A operand instead of global loads.
// =====================================================================
__global__ __launch_bounds__(32)
void crestereo_corr_iter_kernel(const float* __restrict__ fmap1,
                                const float* __restrict__ fmap2,
                                const float* __restrict__ flow,
                                float* __restrict__ out)
{
    const int ntx = WW / TW;                 // 15
    int bid = blockIdx.x;
    int tx = bid % ntx;
    int y  = (bid / ntx) % HH;
    int n  = bid / (ntx * HH);
    int x0 = tx * TW;
    int lane = threadIdx.x;

    __shared__ float rw[3][4][18];           // [dy][chunk-channel][col(-1..16)]
    __shared__ float gd[3][256];             // gram dump: [dy][lane*8 + vgpr]

    // ---- static ownership of the 54 warp pixels (3 rows x 18 cols) ----
    // pixel 0: p = lane (always valid); pixel 1: p = lane + 32 (lanes 0..21)
    const bool has1 = (lane + 32) < 54;

    int p0 = lane;
    int yr0 = p0 / 18, col0 = p0 % 18;
    int yc0 = min(max(y - 1 + yr0, 0), HH - 1);
    int xc0 = min(max(x0 - 1 + col0, 0), WW - 1);
    int a00, a01, a02, a03;  float q00, q01, q02, q03;
    bilin_setup(flow, n, yc0, xc0, a00, a01, a02, a03, q00, q01, q02, q03);

    int p1 = has1 ? (lane + 32) : 0;         // dummy 0 when unused
    int yr1 = p1 / 18, col1 = p1 % 18;
    int yc1 = min(max(y - 1 + yr1, 0), HH - 1);
    int xc1 = min(max(x0 - 1 + col1, 0), WW - 1);
    int a10, a11, a12, a13;  float q10, q11, q12, q13;
    bilin_setup(flow, n, yc1, xc1, a10, a11, a12, a13, q10, q11, q12, q13);

    const float* f1n = fmap1 + (size_t)n * CC * HH * WW;
    const float* f2n = fmap2 + (size_t)n * CC * HH * WW;
    float* oit = out + OUT_OFF1;

    const int half  = lane >> 4;     // 0 or 1
    const int m     = lane & 15;     // 0..15 : pixel / N column
    const int kbase = 2 * half;      // K split: lanes<16 -> K{0,1}, else K{2,3}

    for (int g = 0; g < 4; g++) {
        v8f acc0 = {}, acc1 = {}, acc2 = {};
        float fixL0 = 0.f, fixL1 = 0.f, fixL2 = 0.f;
        float fixR0 = 0.f, fixR1 = 0.f, fixR2 = 0.f;

#pragma unroll 1
        for (int cc = 0; cc < CG / 4; cc++) {        // 16 chunks of 4 channels
            int c = g * CG + cc * 4;
            const float* pc = f2n + (size_t)c * (HH * WW);

            // ---- warp chunk (4 channels, 54 pixels) into LDS ----
#pragma unroll
            for (int k = 0; k < 4; k++) {
                const float* p = pc + (size_t)k * (HH * WW);
                rw[yr0][k][col0] = p[a00] * q00 + p[a01] * q01 +
                                   p[a02] * q02 + p[a03] * q03;
            }
            if (has1) {
#pragma unroll
                for (int k = 0; k < 4; k++) {
                    const float* p = pc + (size_t)k * (HH * WW);
                    rw[yr1][k][col1] = p[a10] * q10 + p[a11] * q11 +
                                       p[a12] * q12 + p[a13] * q13;
                }
            }
            __syncthreads();

            // ---- A: fmap1 row tile, 16x4 fp32 (2 VGPRs/lane) ----
            const float* l0 = f1n + (size_t)(c + kbase) * (HH * WW) + (size_t)y * WW;
            v2f a;
            a.x = l0[x0 + m];
            a.y = l0[(size_t)(HH * WW) + x0 + m];

            // ---- B per dy: warped rows, 4x16 fp32 ----
            v2f b0, b1, b2;
            b0.x = rw[0][kbase][1 + m];  b0.y = rw[0][kbase + 1][1 + m];
            b1.x = rw[1][kbase][1 + m];  b1.y = rw[1][kbase + 1][1 + m];
            b2.x = rw[2][kbase][1 + m];  b2.y = rw[2][kbase + 1][1 + m];

            acc0 = __builtin_amdgcn_wmma_f32_16x16x4_f32(false, a, false, b0,
                                                         (short)0, acc0, false, false);
            acc1 = __builtin_amdgcn_wmma_f32_16x16x4_f32(false, a, false, b1,
                                                         (short)0, acc1, false, false);
            acc2 = __builtin_amdgcn_wmma_f32_16x16x4_f32(false, a, false, b2,
                                                         (short)0, acc2, false, false);

            // ---- halo fixups via lane broadcast of A (no extra loads) ----
            // L[c+0..3, x0]    lives in lanes 0 (K0,K1) and 16 (K2,K3)
            // L[c+0..3, x0+15] lives in lanes 15 (K0,K1) and 31 (K2,K3)
            float lL[4], lR[4];
            lL[0] = __shfl(a.x, 0, 32);   lL[1] = __shfl(a.y, 0, 32);
            lL[2] = __shfl(a.x, 16, 32);  lL[3] = __shfl(a.y, 16, 32);
            lR[0] = __shfl(a.x, 15, 32);  lR[1] = __shfl(a.y, 15, 32);
            lR[2] = __shfl(a.x, 31, 32);  lR[3] = __shfl(a.y, 31, 32);
#pragma unroll
            for (int k = 0; k < 4; k++) {
                fixL0 += lL[k] * rw[0][k][0];   fixR0 += lR[k] * rw[0][k][17];
                fixL1 += lL[k] * rw[1][k][0];   fixR1 += lR[k] * rw[1][k][17];
                fixL2 += lL[k] * rw[2][k][0];   fixR2 += lR[k] * rw[2][k][17];
            }
            __syncthreads();
        }

        // ---- dump grams to LDS for diagonal extraction ----
#pragma unroll
        for (int k = 0; k < 8; k++) {
            gd[0][lane * 8 + k] = acc0[k];
            gd[1][lane * 8 + k] = acc1[k];
            gd[2][lane * 8 + k] = acc2[k];
        }
        __syncthreads();

        if (lane < 16) {
            int i = lane;                                  // output pixel in tile
            float fl[3] = {fixL0, fixL1, fixL2};
            float fr[3] = {fixR0, fixR1, fixR2};
#pragma unroll
            for (int dy = 0; dy < 3; dy++) {
#pragma unroll
                for (int dx = 0; dx < 3; dx++) {
                    int j = i + dx - 1;
                    float v;
                    if (j < 0)        v = fl[dy];
                    else if (j > 15)  v = fr[dy];
                    else              v = gd[dy][(j + 16 * (i >> 3)) * 8 + (i & 7)];
                    int ch = g * SS + dy * 3 + dx;
                    oit[(((size_t)n * 36 + ch) * HH + y) * WW + x0 + i] = v * (1.0f / 64.0f);
                }
            }
        }
        __syncthreads();
    }
}

// =====================================================================
extern "C" void kernel_launch(void* const* d_in, const int* in_sizes, int n_in,
                              void* d_out, int out_size, void* d_ws, size_t ws_size,
                              hipStream_t stream)
{
    const float* fmap1 = (const float*)d_in[0];
    const float* fmap2 = (const float*)d_in[1];
    const float* flow  = (const float*)d_in[2];
    const float* eo    = (const float*)d_in[3];
    const float* mask  = (const float*)d_in[4];
    float* out = (float*)d_out;

    // flow_up
    {
        int total = NN * HH * WW;                 // 61440
        crestereo_upsample_kernel<<<(total + 255) / 256, 256, 0, stream>>>(flow, mask, out);
    }
    // corr_att_offset
    {
        int total = NN * SS * HH * WW;            // 552960
        crestereo_corr_att_kernel<<<(total + 255) / 256, 256, 0, stream>>>(
            fmap1, fmap2, flow, eo, out);
    }
    // corr_iter (WMMA)
    {
        int blocks = NN * HH * (WW / TW);         // 3840 tiles, 1 wave each
        crestereo_corr_iter_kernel<<<blocks, 32, 0, stream>>>(fmap1, fmap2, flow, out);
    }
}